// FractalAttentionCore_34007551050111
// MI455X (gfx1250) — compile-verified
//
#include <hip/hip_runtime.h>
#include <math.h>

typedef __attribute__((ext_vector_type(2))) float v2f;
typedef __attribute__((ext_vector_type(4))) float v4f;
typedef __attribute__((ext_vector_type(8))) float v8f;

#define B_      8
#define H_      1024
#define MAXCAN  160
#define NCHUNK  128

// workspace layout (bytes)
#define WS_NLOC   0
#define WS_NCAN   4
#define WS_LIDX   8                       // 64 ints
#define WS_CIDX   (8 + 64 * 4)            // 160 ints
#define WS_POUT   1024                    // 2 * B * H floats (hilbert-out, cantor-out)
#define WS_VPART  (1024 + 2 * B_ * H_ * 4)// B * NCHUNK * H floats

// ---------------- Hilbert curve helpers (exact integer port of reference) -------------
__device__ inline void hilbert_d2xy(int n, int d, int& x, int& y) {
    x = 0; y = 0;
    for (int s = 1; s < n; s *= 2) {
        int rx = 1 & (d / 2);
        int ry = 1 & (d ^ rx);
        if (ry == 0) {
            if (rx == 1) { x = s - 1 - x; y = s - 1 - y; }
            int t = x; x = y; y = t;
        }
        x += s * rx; y += s * ry;
        d /= 4;
    }
}
__device__ inline int hilbert_xy2d(int n, int x, int y) {
    int d = 0;
    for (int s = n / 2; s > 0; s /= 2) {
        int rx = (x & s) ? 1 : 0;
        int ry = (y & s) ? 1 : 0;
        d += s * s * ((3 * rx) ^ ry);
        if (ry == 0) {
            if (rx == 1) { x = s - 1 - x; y = s - 1 - y; }
            int t = x; x = y; y = t;
        }
    }
    return d;
}

// ---------------- Kernel 1: build Hilbert-window + Cantor index lists ----------------
__global__ void setup_indices_kernel(const int* __restrict__ qidx_p, int L,
                                     int* __restrict__ ws_i) {
    if (threadIdx.x != 0) return;
    // --- Hilbert local window: order = ceil(log2(sqrt(10000))) = 7 -> grid 128 ---
    const int grid = 128;
    const int n_coords = 10000;              // min(MAX_SEQ_LEN, grid*grid)
    int center = qidx_p[0];
    if (center > n_coords - 1) center = n_coords - 1;
    int cx, cy;
    hilbert_d2xy(grid, center, cx, cy);
    int nloc = 0;
    for (int dx = -3; dx <= 3; ++dx) {
        for (int dy = -3; dy <= 3; ++dy) {
            int x = cx + dx, y = cy + dy;
            if (x >= 0 && x < grid && y >= 0 && y < grid) {
                int i = hilbert_xy2d(grid, x, y);
                if (i < L) ws_i[WS_LIDX / 4 + nloc++] = i;
            }
        }
    }
    ws_i[WS_NLOC / 4] = nloc;

    // --- Cantor endpoints: float64 recursion, depth = min(floor(log2 L), 7) ---
    int flog2 = 31 - __clz(L);
    int depth = flog2 < 7 ? flog2 : 7;
    double a[129], bb[129];
    a[0] = 0.0; a[1] = 1.0;
    int len = 2;
    for (int it = 0; it < depth; ++it) {
        for (int i = 0; i < len - 1; ++i) {
            double l = a[i], r = a[i + 1];
            double third = (r - l) / 3.0;
            bb[2 * i] = l; bb[2 * i + 1] = l + third;
        }
        bb[2 * (len - 1)] = a[len - 1];
        len = 2 * (len - 1) + 1;
        for (int i = 0; i < len; ++i) a[i] = bb[i];
    }
    long long ci[129];
    for (int i = 0; i < len; ++i) ci[i] = (long long)(a[i] * (double)(L - 1));
    for (int i = 1; i < len; ++i) {           // insertion sort
        long long v = ci[i]; int j = i - 1;
        while (j >= 0 && ci[j] > v) { ci[j + 1] = ci[j]; --j; }
        ci[j + 1] = v;
    }
    int ncan = 0;
    for (int i = 0; i < len; ++i)
        if (i == 0 || ci[i] != ci[i - 1]) ws_i[WS_CIDX / 4 + ncan++] = (int)ci[i];
    ws_i[WS_NCAN / 4] = ncan;
}

// ---------------- Kernel 2: sparse attention (Hilbert / Cantor) ----------------
// Scores via fp32 WMMA 16x16x4: A = 16-key x 4-H tile of K, B = q broadcast to all
// 16 columns, C accumulated over H/4 steps. All D columns are identical.
__global__ __launch_bounds__(256) void sparse_attn_kernel(
    const float* __restrict__ q, const float* __restrict__ K,
    const float* __restrict__ V, int L, const int* __restrict__ ws_i,
    float* __restrict__ pout) {
    const int b   = blockIdx.x;
    const int pat = blockIdx.y;              // 0 = hilbert, 1 = cantor
    const int tid = threadIdx.x;

    __shared__ float s_scores[MAXCAN];
    __shared__ float s_w[MAXCAN];
    __shared__ int   s_idx[MAXCAN];

    const int  cnt  = ws_i[pat == 0 ? WS_NLOC / 4 : WS_NCAN / 4];
    const int* gidx = ws_i + (pat == 0 ? WS_LIDX / 4 : WS_CIDX / 4);
    if (tid < MAXCAN)
        s_idx[tid] = (tid < cnt) ? gidx[tid] : (cnt > 0 ? gidx[0] : 0);
    __syncthreads();

    const int ntile = (cnt + 15) >> 4;
    if (tid < 32) {                          // wave 0 only: EXEC all-ones for WMMA
        const int   lane = tid;
        const int   m    = lane & 15;        // key row in tile (A: M = lane&15)
        const int   kk   = (lane >> 4) << 1; // A/B layout: half-wave holds K-pair 0-1 / 2-3
        const float scale = 1.0f / 32.0f;    // 1/sqrt(H)
        const float* qrow = q + (size_t)b * H_;
        for (int t = 0; t < ntile; ++t) {
            const int key = s_idx[t * 16 + m];
            const float* krow = K + ((size_t)b * L + key) * H_;
            v8f c = {};
            #pragma unroll 4
            for (int h0 = 0; h0 < H_; h0 += 4) {
                v2f av = *(const v2f*)(krow + h0 + kk);
                v2f bv = *(const v2f*)(qrow + h0 + kk);
                c = __builtin_amdgcn_wmma_f32_16x16x4_f32(
                        false, av, false, bv, (short)0, c, false, false);
            }
            // D[m][n] identical across n; lane 0 holds M=0..7, lane 16 holds M=8..15
            if (lane == 0) {
                #pragma unroll
                for (int r = 0; r < 8; ++r) s_scores[t * 16 + r] = c[r] * scale;
            }
            if (lane == 16) {
                #pragma unroll
                for (int r = 0; r < 8; ++r) s_scores[t * 16 + 8 + r] = c[r] * scale;
            }
        }
    }
    __syncthreads();

    if (tid == 0) {                          // tiny softmax (<=160 entries)
        float mx = -3.402823466e+38f;
        for (int j = 0; j < cnt; ++j) mx = fmaxf(mx, s_scores[j]);
        float sum = 0.f;
        for (int j = 0; j < cnt; ++j) {
            float e = expf(s_scores[j] - mx);
            s_w[j] = e; sum += e;
        }
        float inv = (sum > 0.f) ? 1.0f / sum : 0.f;
        for (int j = 0; j < cnt; ++j) s_w[j] *= inv;
    }
    __syncthreads();

    // weighted V gather: thread t -> H dims [4t, 4t+3], coalesced float4 loads
    const int h = tid * 4;
    v4f acc = {0.f, 0.f, 0.f, 0.f};
    for (int j = 0; j < cnt; ++j) {
        const float* vrow = V + ((size_t)b * L + s_idx[j]) * H_ + h;
        v4f vv = *(const v4f*)vrow;
        acc += s_w[j] * vv;
    }
    *(v4f*)(pout + ((size_t)(pat * B_ + b)) * H_ + h) = acc;
}

// ---------------- Kernel 3: streaming partial sums of V (dragon = mean of V) --------
// dragon weights are identically zero (pattern stays all-0.5; (p-min)/(max-min+eps)=0),
// so dragon attention is a uniform softmax -> row-mean of V. Pure bandwidth:
// 256 MB single-use stream > 192 MB L2, so use nontemporal (NT) loads and keep
// >=8 b128 loads in flight per wave.
__global__ __launch_bounds__(256) void vmean_partial_kernel(
    const float* __restrict__ V, int L, float* __restrict__ part) {
    const int c = blockIdx.x;
    const int b = blockIdx.y;
    const int rows = L / NCHUNK;             // 64 rows per block
    const int h = threadIdx.x * 4;
    const v4f* base = (const v4f*)(V + ((size_t)b * L + (size_t)c * rows) * H_ + h);
    const size_t stride = H_ / 4;            // v4f elements per row
    v4f acc = {0.f, 0.f, 0.f, 0.f};
    #pragma unroll 8
    for (int r = 0; r < rows; ++r)
        acc += __builtin_nontemporal_load(base + (size_t)r * stride);
    v4f* dst = (v4f*)(part + ((size_t)(b * NCHUNK + c)) * H_ + h);
    __builtin_nontemporal_store(acc, dst);
}

// ---------------- Kernel 4: reduce partials + combine with softmax(pattern_weights) --
__global__ __launch_bounds__(256) void combine_kernel(
    const float* __restrict__ pw, const float* __restrict__ pout,
    const float* __restrict__ part, int L, float* __restrict__ out) {
    const int b = blockIdx.x;
    const int h = threadIdx.x * 4;
    const float p0 = pw[0], p1 = pw[1], p2 = pw[2];
    const float m  = fmaxf(p0, fmaxf(p1, p2));
    const float e0 = expf(p0 - m), e1 = expf(p1 - m), e2 = expf(p2 - m);
    const float inv = 1.0f / (e0 + e1 + e2);
    const float w0 = e0 * inv, w1 = e1 * inv, w2 = e2 * inv;

    v4f dragon = {0.f, 0.f, 0.f, 0.f};
    #pragma unroll 8
    for (int c = 0; c < NCHUNK; ++c)
        dragon += *(const v4f*)(part + ((size_t)(b * NCHUNK + c)) * H_ + h);
    dragon *= (1.0f / (float)L);

    v4f lo = *(const v4f*)(pout + ((size_t)(0 * B_ + b)) * H_ + h);
    v4f ca = *(const v4f*)(pout + ((size_t)(1 * B_ + b)) * H_ + h);
    v4f r  = w0 * lo + w1 * ca + w2 * dragon;
    *(v4f*)(out + (size_t)b * H_ + h) = r;
}

extern "C" void kernel_launch(void* const* d_in, const int* in_sizes, int n_in,
                              void* d_out, int out_size, void* d_ws, size_t ws_size,
                              hipStream_t stream) {
    const float* q  = (const float*)d_in[0];   // (B,1,H)
    const float* K  = (const float*)d_in[1];   // (B,L,H)
    const float* V  = (const float*)d_in[2];   // (B,L,H)
    const float* pw = (const float*)d_in[3];   // (3,)
    const int* qidx = (const int*)d_in[4];     // scalar

    const int L = in_sizes[1] / (B_ * H_);     // 8192

    int*   ws_i = (int*)d_ws;
    float* pout = (float*)((char*)d_ws + WS_POUT);
    float* part = (float*)((char*)d_ws + WS_VPART);

    setup_indices_kernel<<<1, 32, 0, stream>>>(qidx, L, ws_i);
    sparse_attn_kernel<<<dim3(B_, 2), 256, 0, stream>>>(q, K, V, L, ws_i, pout);
    vmean_partial_kernel<<<dim3(NCHUNK, B_), 256, 0, stream>>>(V, L, part);
    combine_kernel<<<B_, 256, 0, stream>>>(pw, pout, part, L, (float*)d_out);
}